// STGraphTGCN_1786706395616
// MI455X (gfx1250) — compile-verified
//
#include <hip/hip_runtime.h>
#include <hip/hip_bf16.h>
#include <math.h>

// ---------------------------------------------------------------------------
// TGCN cell on gfx1250 (MI455X). fp32 WMMA (V_WMMA_F32_16X16X4_F32) for all
// GEMMs, TDM (tensor_load_to_lds) weight staging, global fp32 atomics for the
// edge scatter (L2-resident working set: M+agg = 154MB < 192MB L2).
// ---------------------------------------------------------------------------

typedef __attribute__((ext_vector_type(2))) float v2f;
typedef __attribute__((ext_vector_type(8))) float v8f;
typedef __attribute__((ext_vector_type(4))) unsigned int u32x4;
typedef __attribute__((ext_vector_type(4))) int i32x4;
typedef __attribute__((ext_vector_type(8))) int i32x8;

#define NN 100000      // nodes (divisible by 16 -> no tail handling)
#define EE 800000      // edges
#define FF 64          // node feature dim
#define HH 64          // hidden dim
#define H3 192         // 3*H fused gate width

#if defined(__AMDGCN__) && __has_builtin(__builtin_amdgcn_tensor_load_to_lds) && \
    __has_builtin(__builtin_amdgcn_s_wait_tensorcnt)
#define USE_TDM 1
#else
#define USE_TDM 0
#endif

// ---- WMMA f32 16x16x4 helpers ---------------------------------------------
// A (16x4, MxK): lane m (0..15) holds row m; VGPR0/1 = K {0,1} for lanes 0-15,
//                K {2,3} for lanes 16-31.
// B (4x16, KxN): lane n (0..15) holds col n; VGPR0/1 = rows K {0,1} (lanes
//                0-15) / K {2,3} (lanes 16-31).
// C/D (16x16):   VGPR v: lanes 0-15 -> M=v, lanes 16-31 -> M=v+8; N = lane%16.

__device__ __forceinline__ v8f wmma4(v2f a, v2f b, v8f c) {
    return __builtin_amdgcn_wmma_f32_16x16x4_f32(
        /*neg_a=*/false, a, /*neg_b=*/false, b,
        /*c_mod=*/(short)0, c, /*reuse_a=*/false, /*reuse_b=*/false);
}

__device__ __forceinline__ v2f load_a_frag(const float* __restrict__ X, int ldx,
                                           int r0, int k0, int lane) {
    int m  = lane & 15;
    int kh = (lane >> 4) & 1;
    const float* p = X + (size_t)(r0 + m) * ldx + (k0 + 2 * kh);
    v2f a;
    a.x = p[0];
    a.y = p[1];
    return a;
}

__device__ __forceinline__ v2f load_b_frag_lds(const float* W, int ldw,
                                               int k0, int c0, int lane) {
    int n  = lane & 15;
    int kh = (lane >> 4) & 1;
    int k  = k0 + 2 * kh;
    v2f b;
    b.x = W[k * ldw + c0 + n];
    b.y = W[(k + 1) * ldw + c0 + n];
    return b;
}

__device__ __forceinline__ void store_c(float* __restrict__ O, int ldo,
                                        int r0, int c0, int lane, v8f acc) {
    int n  = lane & 15;
    int mo = ((lane >> 4) & 1) * 8;
#pragma unroll
    for (int v = 0; v < 8; ++v)
        O[(size_t)(r0 + mo + v) * ldo + c0 + n] = acc[v];
}

// ---- TDM: stage a contiguous [rows x 64] f32 block into LDS ---------------
// Issued by one wave; caller must __syncthreads() afterwards.
__device__ __forceinline__ void tdm_stage(float* ldsbuf, const float* src,
                                          unsigned int rows) {
#if USE_TDM
    unsigned int lds_off = (unsigned int)(size_t)(void*)ldsbuf;
    unsigned long long ga = (unsigned long long)(size_t)src;
    u32x4 g0;
    g0[0] = 1u;                                         // count=1, user D#
    g0[1] = lds_off;                                    // lds_addr [63:32]
    g0[2] = (unsigned int)(ga & 0xFFFFFFFFu);           // global_addr lo
    g0[3] = (unsigned int)((ga >> 32) & 0x1FFFFFFu)     // global_addr [56:32]
            | (2u << 30);                               // type=2 ("image")
    i32x8 g1;
    g1[0] = (int)(2u << 16);                            // data_size = 4B
    g1[1] = (int)(64u << 16);                           // tensor_dim0 = 64
    g1[2] = (int)(rows << 16);                          // tensor_dim1 = rows
    g1[3] = (int)(64u << 16);                           // tile_dim0 = 64
    g1[4] = (int)rows;                                  // tile_dim1 = rows
    g1[5] = 64;                                         // tensor_dim0_stride
    g1[6] = 0;
    g1[7] = 0;
    i32x4 gz = {0, 0, 0, 0};
#if defined(__clang_major__) && (__clang_major__ >= 23)
    i32x8 gz8 = {0, 0, 0, 0, 0, 0, 0, 0};
    __builtin_amdgcn_tensor_load_to_lds(g0, g1, gz, gz, gz8, 0);
#else
    __builtin_amdgcn_tensor_load_to_lds(g0, g1, gz, gz, 0);
#endif
    __builtin_amdgcn_s_wait_tensorcnt(0);
#else
    (void)ldsbuf; (void)src; (void)rows;
#endif
}

// ---- 1) M = node_feat @ [Wz | Wr | Wh]  ->  ws0 [N,192] -------------------
__global__ void __launch_bounds__(256)
k_xform(const float* __restrict__ X,
        const float* __restrict__ Wz, const float* __restrict__ Wr,
        const float* __restrict__ Wh,
        float* __restrict__ M) {
    __shared__ float Wl[64 * H3];                      // 48 KB (interleaved)
    int tid = threadIdx.x;
    for (int i = tid; i < 64 * 64; i += 256) {
        int r = i >> 6, c = i & 63;
        Wl[r * H3 + c]       = Wz[i];
        Wl[r * H3 + 64 + c]  = Wr[i];
        Wl[r * H3 + 128 + c] = Wh[i];
    }
    __syncthreads();

    int lane = tid & 31, wave = tid >> 5;
    int r0 = (blockIdx.x * 8 + wave) * 16;
    if (r0 >= NN) return;                              // wave-uniform

    v2f a[16];                                         // 16x64 A tile, reused
#pragma unroll
    for (int kk = 0; kk < 16; ++kk) a[kk] = load_a_frag(X, FF, r0, kk * 4, lane);

    for (int ct = 0; ct < 12; ++ct) {                  // 12 x 16 output cols
        v8f acc = {};
#pragma unroll
        for (int kk = 0; kk < 16; ++kk)
            acc = wmma4(a[kk], load_b_frag_lds(Wl, H3, kk * 4, ct * 16, lane), acc);
        store_c(M, H3, r0, ct * 16, lane, acc);
    }
}

// ---- 2) agg = broadcast biases [bz|br|bh] ---------------------------------
__global__ void __launch_bounds__(256)
k_init_bias(const float* __restrict__ bz, const float* __restrict__ br,
            const float* __restrict__ bh, float* __restrict__ agg) {
    size_t i = (size_t)blockIdx.x * 256 + threadIdx.x;
    if (i >= (size_t)NN * H3) return;
    int c = (int)(i % H3);
    agg[i] = (c < 64) ? bz[c] : (c < 128) ? br[c - 64] : bh[c - 128];
}

// ---- 3) agg[dst] += w_e * M[src]  (fused z|r|h, one wave per edge) --------
__global__ void __launch_bounds__(256)
k_scatter(const long long* __restrict__ eidx, const float* __restrict__ ew,
          const float* __restrict__ M, float* __restrict__ agg) {
    int e = blockIdx.x * 8 + (threadIdx.x >> 5);
    if (e >= EE) return;
    int lane = threadIdx.x & 31;
    long long s = eidx[e];
    long long d = eidx[(size_t)EE + e];
    float w = ew[e];
    const float4* ms4 = (const float4*)(M + (size_t)s * H3);
    float* ad = agg + (size_t)d * H3;
    __builtin_prefetch(ad + lane * 4, 0, 1);           // warm L2 for atomics
#pragma unroll
    for (int j = 0; j < 2; ++j) {                      // 48 float4 per edge
        int c4 = j * 32 + lane;
        if (c4 < 48) {
            float4 v = ms4[c4];
            float* p = ad + c4 * 4;
            atomicAdd(p + 0, w * v.x);
            atomicAdd(p + 1, w * v.y);
            atomicAdd(p + 2, w * v.z);
            atomicAdd(p + 3, w * v.w);
        }
    }
}

// ---- 4) U_g = agg_g @ Wl_g[:64] + h @ Wl_g[64:] + bl_g  (grid.y = gate) ---
__global__ void __launch_bounds__(256)
k_gates(const float* __restrict__ AGG, const float* __restrict__ Hst,
        const float* __restrict__ Wlz, const float* __restrict__ Wlr,
        const float* __restrict__ Wlh,
        const float* __restrict__ blz, const float* __restrict__ blr,
        const float* __restrict__ blh,
        float* __restrict__ U) {
    __shared__ float Wl[128 * 64];                     // 32 KB
    int g = blockIdx.y;
    const float* Wg = (g == 0) ? Wlz : (g == 1) ? Wlr : Wlh;
    const float* bg = (g == 0) ? blz : (g == 1) ? blr : blh;
    int tid = threadIdx.x;
#if USE_TDM
    if ((tid >> 5) == 0) tdm_stage(Wl, Wg, 128u);
#else
    for (int i = tid; i < 128 * 64; i += 256) Wl[i] = Wg[i];
#endif
    __syncthreads();

    int lane = tid & 31, wave = tid >> 5;
    int r0 = (blockIdx.x * 8 + wave) * 16;
    if (r0 >= NN) return;

    v2f a0[16], a1[16];
#pragma unroll
    for (int kk = 0; kk < 16; ++kk) {
        a0[kk] = load_a_frag(AGG, H3, r0, g * 64 + kk * 4, lane);  // c_g
        a1[kk] = load_a_frag(Hst, HH, r0, kk * 4, lane);           // h
    }
    int n = lane & 15;
    for (int ct = 0; ct < 4; ++ct) {
        float bv = bg[ct * 16 + n];
        v8f acc = {bv, bv, bv, bv, bv, bv, bv, bv};
#pragma unroll
        for (int kk = 0; kk < 16; ++kk)
            acc = wmma4(a0[kk], load_b_frag_lds(Wl, 64, kk * 4, ct * 16, lane), acc);
#pragma unroll
        for (int kk = 0; kk < 16; ++kk)
            acc = wmma4(a1[kk], load_b_frag_lds(Wl + 64 * 64, 64, kk * 4, ct * 16, lane), acc);
        store_c(U, H3, r0, g * 64 + ct * 16, lane, acc);
    }
}

// ---- 5) z = sigmoid(uz); s = h * sigmoid(ur) ------------------------------
__global__ void __launch_bounds__(256)
k_act(const float* __restrict__ U, const float* __restrict__ Hst,
      float* __restrict__ ZS) {
    size_t i = (size_t)blockIdx.x * 256 + threadIdx.x;
    if (i >= (size_t)NN * HH) return;
    size_t row = i >> 6;
    int c = (int)(i & 63);
    float uz = U[row * H3 + c];
    float ur = U[row * H3 + 64 + c];
    float z = 1.f / (1.f + __expf(-uz));
    float r = 1.f / (1.f + __expf(-ur));
    ZS[row * H3 + c]      = z;
    ZS[row * H3 + 64 + c] = Hst[i] * r;
}

// ---- 6) h~ = tanh(uh1 + s @ Wlh[64:]); h' = z*h + (1-z)*h~ ----------------
__global__ void __launch_bounds__(256)
k_candidate(const float* __restrict__ U, const float* __restrict__ ZS,
            const float* __restrict__ Hst, const float* __restrict__ Wlh,
            float* __restrict__ Out) {
    __shared__ float Wb[64 * 64];                      // bottom half of Wl_h
    int tid = threadIdx.x;
#if USE_TDM
    if ((tid >> 5) == 0) tdm_stage(Wb, Wlh + 64 * 64, 64u);
#else
    for (int i = tid; i < 64 * 64; i += 256) Wb[i] = Wlh[64 * 64 + i];
#endif
    __syncthreads();

    int lane = tid & 31, wave = tid >> 5;
    int r0 = (blockIdx.x * 8 + wave) * 16;
    if (r0 >= NN) return;

    v2f a[16];
#pragma unroll
    for (int kk = 0; kk < 16; ++kk)
        a[kk] = load_a_frag(ZS, H3, r0, 64 + kk * 4, lane);        // s = h*r
    int n = lane & 15;
    int mo = ((lane >> 4) & 1) * 8;
    float* Hout = Out + (size_t)NN * FF;               // h' half of d_out
    for (int ct = 0; ct < 4; ++ct) {
        v8f acc;
#pragma unroll
        for (int v = 0; v < 8; ++v)                    // acc = uh1 (C layout)
            acc[v] = U[(size_t)(r0 + mo + v) * H3 + 128 + ct * 16 + n];
#pragma unroll
        for (int kk = 0; kk < 16; ++kk)
            acc = wmma4(a[kk], load_b_frag_lds(Wb, 64, kk * 4, ct * 16, lane), acc);
#pragma unroll
        for (int v = 0; v < 8; ++v) {
            size_t row = (size_t)(r0 + mo + v);
            int col = ct * 16 + n;
            float ht = tanhf(acc[v]);
            float z  = ZS[row * H3 + col];
            float h  = Hst[row * HH + col];
            Hout[row * HH + col] = z * h + (1.f - z) * ht;
        }
    }
}

// ---- 7) y = relu(h') @ W_out + b_out --------------------------------------
__global__ void __launch_bounds__(256)
k_readout(const float* __restrict__ Wout, const float* __restrict__ bout,
          float* __restrict__ Out) {
    __shared__ float Wl[64 * 64];
    int tid = threadIdx.x;
#if USE_TDM
    if ((tid >> 5) == 0) tdm_stage(Wl, Wout, 64u);
#else
    for (int i = tid; i < 64 * 64; i += 256) Wl[i] = Wout[i];
#endif
    __syncthreads();

    int lane = tid & 31, wave = tid >> 5;
    int r0 = (blockIdx.x * 8 + wave) * 16;
    if (r0 >= NN) return;

    const float* Hn = Out + (size_t)NN * FF;
    int m = lane & 15, kh = (lane >> 4) & 1;
    v2f a[16];
#pragma unroll
    for (int kk = 0; kk < 16; ++kk) {                  // relu on load
        const float* p = Hn + (size_t)(r0 + m) * HH + kk * 4 + 2 * kh;
        v2f t;
        t.x = fmaxf(p[0], 0.f);
        t.y = fmaxf(p[1], 0.f);
        a[kk] = t;
    }
    int n = lane & 15;
    for (int ct = 0; ct < 4; ++ct) {
        float bv = bout[ct * 16 + n];
        v8f acc = {bv, bv, bv, bv, bv, bv, bv, bv};
#pragma unroll
        for (int kk = 0; kk < 16; ++kk)
            acc = wmma4(a[kk], load_b_frag_lds(Wl, 64, kk * 4, ct * 16, lane), acc);
        store_c(Out, FF, r0, ct * 16, lane, acc);      // y half of d_out
    }
}

// ---------------------------------------------------------------------------
extern "C" void kernel_launch(void* const* d_in, const int* in_sizes, int n_in,
                              void* d_out, int out_size, void* d_ws, size_t ws_size,
                              hipStream_t stream) {
    const float* node_feat = (const float*)d_in[0];
    const float* edge_w    = (const float*)d_in[1];
    const float* hidden    = (const float*)d_in[2];
    const float* W_z  = (const float*)d_in[3];
    const float* b_z  = (const float*)d_in[4];
    const float* W_r  = (const float*)d_in[5];
    const float* b_r  = (const float*)d_in[6];
    const float* W_h  = (const float*)d_in[7];
    const float* b_h  = (const float*)d_in[8];
    const float* Wl_z = (const float*)d_in[9];
    const float* bl_z = (const float*)d_in[10];
    const float* Wl_r = (const float*)d_in[11];
    const float* bl_r = (const float*)d_in[12];
    const float* Wl_h = (const float*)d_in[13];
    const float* bl_h = (const float*)d_in[14];
    const float* W_out = (const float*)d_in[15];
    const float* b_out = (const float*)d_in[16];
    const long long* eidx = (const long long*)d_in[17];   // int64 [2,E]

    float* out = (float*)d_out;                  // [N*64 y][N*64 h]
    float* M   = (float*)d_ws;                   // [N,192]; reused as U
    float* AGG = M + (size_t)NN * H3;            // [N,192]; reused as ZS
    float* U   = M;
    float* ZS  = AGG;

    const int tile_blocks = (NN / 16 + 7) / 8;   // 782 blocks of 8 waves

    k_xform<<<tile_blocks, 256, 0, stream>>>(node_feat, W_z, W_r, W_h, M);

    const size_t nb_bias = ((size_t)NN * H3 + 255) / 256;
    k_init_bias<<<(int)nb_bias, 256, 0, stream>>>(b_z, b_r, b_h, AGG);

    k_scatter<<<(EE + 7) / 8, 256, 0, stream>>>(eidx, edge_w, M, AGG);

    dim3 ggrid(tile_blocks, 3, 1);
    k_gates<<<ggrid, 256, 0, stream>>>(AGG, hidden, Wl_z, Wl_r, Wl_h,
                                       bl_z, bl_r, bl_h, U);

    const size_t nb_act = ((size_t)NN * HH + 255) / 256;
    k_act<<<(int)nb_act, 256, 0, stream>>>(U, hidden, ZS);

    k_candidate<<<tile_blocks, 256, 0, stream>>>(U, ZS, hidden, Wl_h, out);

    k_readout<<<tile_blocks, 256, 0, stream>>>(W_out, b_out, out);
}